// InterClusterGAT_1666447311293
// MI455X (gfx1250) — compile-verified
//
#include <hip/hip_runtime.h>
#include <hip/hip_bf16.h>

// Problem constants (match the reference).
#define N_VAR    100000
#define D_IN     512
#define D_OUT    512
#define C_CLUST  4096
#define K_MEMB   32
#define E_EDGE   16384
#define S_SHARE  16
#define H_HEADS  4
#define NEG_SLOPE 0.2f

typedef __attribute__((ext_vector_type(2))) float v2f;
typedef __attribute__((ext_vector_type(8))) float v8f;

// -------------------------------------------------------------------------
// Kernel 1: per-cluster mean pooling.  feats[c,:] = mean_k x[ids[c,k],:]
// One block per cluster, 128 threads, float4 per thread (512 dims).
// Member rows are 2KB contiguous -> coalesced global_load_b128.
// -------------------------------------------------------------------------
__global__ void pool_kernel(const float* __restrict__ x,
                            const int* __restrict__ ids,
                            float* __restrict__ feats) {
  const int c  = blockIdx.x;
  const int d0 = threadIdx.x * 4;
  const int* cid = ids + c * K_MEMB;          // uniform -> scalar loads
  float a0 = 0.f, a1 = 0.f, a2 = 0.f, a3 = 0.f;
#pragma unroll 8
  for (int k = 0; k < K_MEMB; ++k) {
    const float4 v = *reinterpret_cast<const float4*>(
        x + (size_t)cid[k] * D_IN + d0);
    a0 += v.x; a1 += v.y; a2 += v.z; a3 += v.w;
  }
  float4 r;
  r.x = a0 * (1.0f / K_MEMB); r.y = a1 * (1.0f / K_MEMB);
  r.z = a2 * (1.0f / K_MEMB); r.w = a3 * (1.0f / K_MEMB);
  *reinterpret_cast<float4*>(feats + (size_t)c * D_IN + d0) = r;
}

// -------------------------------------------------------------------------
// Kernel 2: fused Q/K/V GEMM with V_WMMA_F32_16X16X4_F32 (full fp32).
//   Q = feats @ W_Q^T  => A[m][k] = feats[m][k],  B[k][n] = W_Q[n][k]
// One wave computes a 32(M) x 16(N) tile for all three weight matrices:
// per K-step it loads 2 A fragments + 3 B fragments and issues 6 WMMAs
// (0.83 loads per WMMA; weight fragments reused across both M sub-tiles).
//
// ISA 32-bit A 16x4 layout: lanes 0-15 (M=lane) hold {K+0,K+1} in
// VGPR0/1; lanes 16-31 hold {K+2,K+3}.  B (4x16) mirrors that with N
// striped across lanes.  Both fragments are a contiguous float2 load.
// C/D layout: VGPR i -> row i (lanes 0-15) / row i+8 (lanes 16-31).
// -------------------------------------------------------------------------
__device__ __forceinline__ v2f ld2(const float* p) {
  return *reinterpret_cast<const v2f*>(p);
}

__device__ __forceinline__ v8f wmma_f32(v2f a, v2f b, v8f c) {
  return __builtin_amdgcn_wmma_f32_16x16x4_f32(
      false, a, false, b, (short)0, c, false, false);
}

__global__ void qkv_wmma_kernel(const float* __restrict__ feats,
                                const float* __restrict__ WQ,
                                const float* __restrict__ WK,
                                const float* __restrict__ WV,
                                float* __restrict__ Q,
                                float* __restrict__ Kc,
                                float* __restrict__ V) {
  const int lane  = threadIdx.x & 31;
  const int wave  = threadIdx.x >> 5;
  const int mBase = blockIdx.x * 32;               // cluster tile (2 x 16)
  const int nBase = blockIdx.y * 64 + wave * 16;   // output-dim tile
  const int r     = lane & 15;          // A: row M ; B: column N
  const int kOff  = (lane >> 4) * 2;    // upper half-wave takes K+2,K+3

  const float* aRow0 = feats + (size_t)(mBase +      r) * D_IN + kOff;
  const float* aRow1 = feats + (size_t)(mBase + 16 + r) * D_IN + kOff;
  const float* qRow  = WQ    + (size_t)(nBase +      r) * D_IN + kOff;
  const float* kRow  = WK    + (size_t)(nBase +      r) * D_IN + kOff;
  const float* vRow  = WV    + (size_t)(nBase +      r) * D_IN + kOff;

  v8f accQ0 = {}; v8f accK0 = {}; v8f accV0 = {};
  v8f accQ1 = {}; v8f accK1 = {}; v8f accV1 = {};
#pragma unroll 2
  for (int k = 0; k < D_IN; k += 4) {
    const v2f a0 = ld2(aRow0 + k);
    const v2f a1 = ld2(aRow1 + k);
    const v2f bq = ld2(qRow + k);
    const v2f bk = ld2(kRow + k);
    const v2f bv = ld2(vRow + k);
    accQ0 = wmma_f32(a0, bq, accQ0);
    accQ1 = wmma_f32(a1, bq, accQ1);
    accK0 = wmma_f32(a0, bk, accK0);
    accK1 = wmma_f32(a1, bk, accK1);
    accV0 = wmma_f32(a0, bv, accV0);
    accV1 = wmma_f32(a1, bv, accV1);
  }

  const int mOut = mBase + (lane >> 4) * 8;
  const int nOut = nBase + r;
#pragma unroll
  for (int i = 0; i < 8; ++i) {
    const size_t o0 = (size_t)(mOut + i) * D_OUT + nOut;
    const size_t o1 = (size_t)(mOut + 16 + i) * D_OUT + nOut;
    Q[o0]  = accQ0[i];  Q[o1]  = accQ1[i];
    Kc[o0] = accK0[i];  Kc[o1] = accK1[i];
    V[o0]  = accV0[i];  V[o1]  = accV1[i];
  }
}

// -------------------------------------------------------------------------
// Kernel 3: per-edge scalar attention.  One wave32 per edge.
//   s = dot(Q[c1], K[c2]) / sqrt(512); leaky_relu; sigmoid * mean(head_w)
// -------------------------------------------------------------------------
__global__ void attn_kernel(const float* __restrict__ Q,
                            const float* __restrict__ Kc,
                            const int* __restrict__ edge_idx,
                            const float* __restrict__ head_w,
                            const int* __restrict__ active_heads_p,
                            float* __restrict__ attn) {
  const int e    = blockIdx.x * 8 + (threadIdx.x >> 5);
  const int lane = threadIdx.x & 31;
  const int c1 = edge_idx[e];
  const int c2 = edge_idx[E_EDGE + e];
  const float* q = Q  + (size_t)c1 * D_OUT + lane * 16;
  const float* k = Kc + (size_t)c2 * D_OUT + lane * 16;
  float s = 0.f;
#pragma unroll
  for (int d = 0; d < 16; d += 4) {
    const float4 qv = *reinterpret_cast<const float4*>(q + d);
    const float4 kv = *reinterpret_cast<const float4*>(k + d);
    s += qv.x * kv.x + qv.y * kv.y + qv.z * kv.z + qv.w * kv.w;
  }
#pragma unroll
  for (int off = 16; off > 0; off >>= 1) s += __shfl_down(s, off, 32);
  if (lane == 0) {
    s *= 0.04419417382415922f;                 // 1/sqrt(512)
    s  = (s >= 0.f) ? s : NEG_SLOPE * s;       // leaky_relu
    const int ah = active_heads_p[0];
    float hw = 0.f;
    for (int h = 0; h < ah; ++h) hw += head_w[h];
    hw /= (float)ah;
    attn[e] = hw / (1.f + __expf(-s));         // sigmoid * mean(head_w)
  }
}

// -------------------------------------------------------------------------
// Kernel 4: x_out = x_var  (float4 grid-stride streaming copy)
// -------------------------------------------------------------------------
__global__ void copy_kernel(const float4* __restrict__ src,
                            float4* __restrict__ dst, int n4) {
  int i = blockIdx.x * blockDim.x + threadIdx.x;
  const int stride = gridDim.x * blockDim.x;
  for (; i < n4; i += stride) dst[i] = src[i];
}

// -------------------------------------------------------------------------
// Kernel 5: scatter-add.  One block per edge; each thread keeps 2 dims of
// upd = attn[e] * V[c2] in registers and atomically adds them into the 16
// shared-var rows (global_atomic_add_f32, no-return / STOREcnt path).
// -------------------------------------------------------------------------
__global__ void scatter_kernel(const float* __restrict__ V,
                               const float* __restrict__ attn,
                               const int* __restrict__ edge_idx,
                               const int* __restrict__ shared_vars,
                               float* __restrict__ out) {
  const int e  = blockIdx.x;
  const int d0 = threadIdx.x * 2;
  const float a  = attn[e];
  const int   c2 = edge_idx[E_EDGE + e];
  const float2 v = *reinterpret_cast<const float2*>(
      V + (size_t)c2 * D_OUT + d0);
  const float u0 = a * v.x;
  const float u1 = a * v.y;
  const int* sv = shared_vars + e * S_SHARE;   // uniform -> scalar loads
#pragma unroll
  for (int s = 0; s < S_SHARE; ++s) {
    float* p = out + (size_t)sv[s] * D_OUT + d0;
    atomicAdd(p,     u0);
    atomicAdd(p + 1, u1);
  }
}

// -------------------------------------------------------------------------
// Host-side launcher.
// Workspace layout (needs ~33.6 MB):
//   [0,8MB)      cluster_feats [4096 x 512] f32
//   [8,16MB)     Q             [4096 x 512] f32
//   [16,24MB)    K             [4096 x 512] f32
//   [24,32MB)    V             [4096 x 512] f32
//   [32MB,+64KB) attn          [16384] f32
// -------------------------------------------------------------------------
extern "C" void kernel_launch(void* const* d_in, const int* in_sizes, int n_in,
                              void* d_out, int out_size, void* d_ws,
                              size_t ws_size, hipStream_t stream) {
  (void)in_sizes; (void)n_in; (void)out_size; (void)ws_size;

  const float* x_var   = (const float*)d_in[0];
  const float* W_Q     = (const float*)d_in[1];
  const float* W_K     = (const float*)d_in[2];
  const float* W_V     = (const float*)d_in[3];
  const float* head_w  = (const float*)d_in[4];
  const int* cluster_var_ids   = (const int*)d_in[5];
  const int* cluster_edge_idx  = (const int*)d_in[6];
  const int* shared_vars       = (const int*)d_in[7];
  const int* active_heads      = (const int*)d_in[8];
  float* out = (float*)d_out;

  char* ws = (char*)d_ws;
  float* feats = (float*)(ws);
  float* Q     = (float*)(ws + ((size_t)8  << 20));
  float* Kc    = (float*)(ws + ((size_t)16 << 20));
  float* V     = (float*)(ws + ((size_t)24 << 20));
  float* attn  = (float*)(ws + ((size_t)32 << 20));

  // 1) cluster mean pooling
  pool_kernel<<<C_CLUST, 128, 0, stream>>>(x_var, cluster_var_ids, feats);

  // 2) fused Q/K/V projection via f32 WMMA (32x16 tile per wave)
  qkv_wmma_kernel<<<dim3(C_CLUST / 32, D_OUT / 64), 128, 0, stream>>>(
      feats, W_Q, W_K, W_V, Q, Kc, V);

  // 3) per-edge attention scalars
  attn_kernel<<<E_EDGE / 8, 256, 0, stream>>>(Q, Kc, cluster_edge_idx,
                                              head_w, active_heads, attn);

  // 4) out = x_var
  copy_kernel<<<2048, 256, 0, stream>>>((const float4*)x_var, (float4*)out,
                                        (N_VAR * D_IN) / 4);

  // 5) scatter-add edge messages into shared vars
  scatter_kernel<<<E_EDGE, 256, 0, stream>>>(V, attn, cluster_edge_idx,
                                             shared_vars, out);
}